// TorchSpanEmo_18433999635168
// MI455X (gfx1250) — compile-verified
//
#include <hip/hip_runtime.h>
#include <hip/hip_bf16.h>
#include <math.h>

// Problem constants (from the reference)
#define B_  32
#define S_  2048
#define E_  1024
#define L_  11
#define V_  30522
#define M_  (B_ * L_)          // 352 rows actually needed
#define MTILES (M_ / 16)       // 22, exact
#define NTILES (E_ / 16)       // 64

typedef __attribute__((ext_vector_type(2))) float v2f;
typedef __attribute__((ext_vector_type(8))) float v8f;

// ---------------------------------------------------------------------------
// Kernel A: for the 352 gathered rows, compute per-(row, n-tile) partial sums
//   partials[m * 64 + ntile] = sum_{j in ntile} tanh(x_m . W1[:,j] + b1[j]) * W2[j]
// using V_WMMA_F32_16X16X4_F32 for the inner 16x16x1024 GEMM tiles.
// Grid: (22, 8) blocks of 256 threads (8 waves); wave w of block (mt, by)
// owns output tile (mt, by*8 + w).
// ---------------------------------------------------------------------------
__global__ __launch_bounds__(256)
void spanemo_gemm_wmma(const int* __restrict__ input_ids,
                       const int* __restrict__ label_idxs,
                       const float* __restrict__ emb,   // (V, E)
                       const float* __restrict__ W1,    // (E, E) row-major [k, j]
                       const float* __restrict__ b1,    // (E)
                       const float* __restrict__ W2,    // (E)
                       float* __restrict__ partials)    // (M_, NTILES)
{
    const int lane  = threadIdx.x & 31;
    const int wave  = threadIdx.x >> 5;
    const int mtile = blockIdx.x;                 // 0..21
    const int ntile = blockIdx.y * 8 + wave;      // 0..63
    const int n0    = ntile * 16;

    // ---- A operand addressing: 16x4 f32 tile ----
    // lanes 0-15: M=lane,    K pair {0,1}; lanes 16-31: M=lane-16, K pair {2,3}
    const int mloc = lane & 15;
    const int m    = mtile * 16 + mloc;           // 0..351 (exact, no masking)
    const int bidx = m / L_;
    const int lidx = m - bidx * L_;
    const int row  = input_ids[bidx * S_ + label_idxs[lidx]];
    const int kb   = (lane >> 4) * 2;             // 0 or 2
    const float* aptr = emb + (size_t)row * E_ + kb;

    // ---- B operand addressing: 4x16 f32 tile ----
    // lanes 0-15: N=lane, K pair {0,1}; lanes 16-31: N=lane-16, K pair {2,3}
    const int col = n0 + (lane & 15);
    const float* bptr = W1 + (size_t)kb * E_ + col;

    v8f c = {};
    #pragma unroll 8
    for (int k = 0; k < E_; k += 4) {
        const float2 av = *(const float2*)(aptr + k);       // 8B aligned
        v2f a; a.x = av.x; a.y = av.y;
        v2f b;
        b.x = bptr[(size_t)k * E_];
        b.y = bptr[(size_t)(k + 1) * E_];
        // D = A(16x4) * B(4x16) + C   (full f32 precision)
        c = __builtin_amdgcn_wmma_f32_16x16x4_f32(
                /*neg_a=*/false, a, /*neg_b=*/false, b,
                /*c_mod=*/(short)0, c, /*reuse_a=*/false, /*reuse_b=*/false);
    }

    // Epilogue: h = tanh(c + b1[col]); partial = sum_cols h * W2[col]
    // C/D layout: c[r] is (row = r + 8*(lane>=16), col = n0 + (lane&15))
    const float b1v = b1[col];
    const float w2v = W2[col];
    float sums[8];
    #pragma unroll
    for (int r = 0; r < 8; ++r) {
        float h = tanhf(c[r] + b1v) * w2v;
        // reduce over the 16 columns held by this half-wave
        h += __shfl_xor(h, 1, 32);
        h += __shfl_xor(h, 2, 32);
        h += __shfl_xor(h, 4, 32);
        h += __shfl_xor(h, 8, 32);
        sums[r] = h;
    }
    if ((lane & 15) == 0) {
        const int rbase = (lane >> 4) * 8;        // lane 0 -> rows 0..7, lane 16 -> 8..15
        #pragma unroll
        for (int r = 0; r < 8; ++r) {
            const int mm = mtile * 16 + rbase + r;
            partials[mm * NTILES + ntile] = sums[r];
        }
    }
}

// ---------------------------------------------------------------------------
// Kernel B: single workgroup. Sum partials -> logits, then loss terms.
// d_out layout: [loss, logits(352), y_pred(352)]
// ---------------------------------------------------------------------------
__global__ __launch_bounds__(512)
void spanemo_finalize(const float* __restrict__ partials,
                      const float* __restrict__ targets,   // (B, L)
                      const float* __restrict__ b2,        // (1)
                      float* __restrict__ out)
{
    __shared__ float logits_s[M_];
    __shared__ float bce_s[512];
    __shared__ float corr_s[B_];

    const int t = threadIdx.x;
    float bce = 0.0f;
    if (t < M_) {
        float s = 0.0f;
        #pragma unroll 8
        for (int j = 0; j < NTILES; ++j) s += partials[t * NTILES + j];
        const float logit = s + b2[0];
        logits_s[t] = logit;
        out[1 + t]  = logit;
        const float yh = 1.0f / (1.0f + expf(-logit));
        out[1 + M_ + t] = (yh > 0.5f) ? 1.0f : 0.0f;
        const float tt = targets[t];
        bce = fmaxf(logit, 0.0f) - logit * tt + log1pf(expf(-fabsf(logit)));
    }
    bce_s[t] = bce;
    __syncthreads();
    for (int s = 256; s > 0; s >>= 1) {
        if (t < s) bce_s[t] += bce_s[t + s];
        __syncthreads();
    }

    if (t < B_) {
        float sz = 0.0f, so = 0.0f;
        int nz = 0, no = 0;
        #pragma unroll
        for (int l = 0; l < L_; ++l) {
            const float logit = logits_s[t * L_ + l];
            const float yh = 1.0f / (1.0f + expf(-logit));
            const float tt = targets[t * L_ + l];
            if (tt > 0.5f) { so += expf(-yh); ++no; }
            else           { sz += expf( yh); ++nz; }
        }
        const float num = (float)nz * (float)no;
        corr_s[t] = (num > 0.0f) ? (sz * so / fmaxf(num, 1.0f)) : 0.0f;
    }
    __syncthreads();
    if (t == 0) {
        float cl = 0.0f;
        #pragma unroll
        for (int i = 0; i < B_; ++i) cl += corr_s[i];
        cl *= (1.0f / (float)B_);
        const float cel = bce_s[0] * (1.0f / (float)M_);
        out[0] = 0.8f * cel + 0.2f * cl;   // (1-ALPHA)*cel + ALPHA*cl
    }
}

extern "C" void kernel_launch(void* const* d_in, const int* in_sizes, int n_in,
                              void* d_out, int out_size, void* d_ws, size_t ws_size,
                              hipStream_t stream) {
    const int*   input_ids  = (const int*)  d_in[0];   // (B, S) int32
    const int*   label_idxs = (const int*)  d_in[1];   // (L,)   int32
    const float* targets    = (const float*)d_in[2];   // (B, L)
    const float* emb        = (const float*)d_in[3];   // (V, E)
    const float* W1         = (const float*)d_in[4];   // (E, E)
    const float* b1         = (const float*)d_in[5];   // (E,)
    const float* W2         = (const float*)d_in[6];   // (E, 1)
    const float* b2         = (const float*)d_in[7];   // (1,)
    float* out      = (float*)d_out;
    float* partials = (float*)d_ws;                    // M_ * NTILES floats (88 KB)

    dim3 grid(MTILES, 8);                              // 22 x 8 blocks
    spanemo_gemm_wmma<<<grid, 256, 0, stream>>>(input_ids, label_idxs, emb, W1,
                                                b1, W2, partials);
    spanemo_finalize<<<1, 512, 0, stream>>>(partials, targets, b2, out);
}